// SINDyModel_40759239639422
// MI455X (gfx1250) — compile-verified
//
#include <hip/hip_runtime.h>
#include <math.h>

typedef __attribute__((ext_vector_type(2))) float v2f;
typedef __attribute__((ext_vector_type(4))) float v4f;
typedef __attribute__((ext_vector_type(8))) float v8f;

#define Bn 16
#define Tn 2048
#define Fn 128
#define Ln 16
#define LIBn 1016

// ---- output layout (flat f32, reference tuple order) ----
static constexpr size_t O_YHAT = 0;
static constexpr size_t O_XHAT = (size_t)Bn * Tn * Ln;              // 524288
static constexpr size_t O_Z    = O_XHAT + (size_t)Bn * Tn * Fn;     // 4718592
static constexpr size_t O_JAC  = O_Z + (size_t)Bn * Tn * Ln;        // 5242880
static constexpr size_t O_WSIN = O_JAC + (size_t)Bn * Tn * Ln * Fn; // 72351744
static constexpr size_t O_WDEC = O_WSIN + (size_t)Ln * LIBn;        // 72368000

// ---- workspace layout (floats) ----
static constexpr size_t W_IM    = 0;                       // B*T*L imag(analytic)
static constexpr size_t W_KTAB  = (size_t)Bn * Tn * Ln;    // 2048-entry cot kernel
static constexpr size_t W_S     = W_KTAB + Tn;             // parity sums [B*L][2]
static constexpr size_t W_COMBO = W_S + (size_t)Bn * Ln * 2; // 1016 packed uint32

__device__ __forceinline__ v8f wmma4(v2f a, v2f b, v8f c) {
  // D(16x16,f32) = A(16x4,f32) x B(4x16,f32) + C
  return __builtin_amdgcn_wmma_f32_16x16x4_f32(false, a, false, b, (short)0, c,
                                               false, false);
}

// ============ K0: build cot-kernel table + monomial index table ============
// ktab[d] = (2/N)*cot(pi*d/N) for odd d, else 0  (Im part of analytic kernel)
// combo[c] packs (i, j, k, src): theta[c] = base[src][i] * z[j] * z[k],
// with index 16 pointing at a 1.0 sentinel column.
__global__ void k_setup(float* __restrict__ ktab, unsigned* __restrict__ combo) {
  int gid = blockIdx.x * blockDim.x + threadIdx.x;
  if (gid < Tn) {
    float v = 0.0f;
    if (gid & 1) {
      float ang = 3.14159265358979323846f * (float)gid / (float)Tn;
      v = (2.0f / (float)Tn) * (cosf(ang) / sinf(ang));
    }
    ktab[gid] = v;
  }
  if (gid == 0) {
    int idx = 0;
    const unsigned SENT = 16u;
    for (unsigned i = 0; i < 16; ++i)                       // degree 1 (= z)
      combo[idx++] = i | (SENT << 5) | (SENT << 10);
    for (unsigned i = 0; i < 16; ++i)                       // degree 2
      for (unsigned j = i; j < 16; ++j)
        combo[idx++] = i | (j << 5) | (SENT << 10);
    for (unsigned i = 0; i < 16; ++i)                       // degree 3
      for (unsigned j = i; j < 16; ++j)
        for (unsigned k = j; k < 16; ++k)
          combo[idx++] = i | (j << 5) | (k << 10);
    for (unsigned i = 0; i < 16; ++i)                       // z again
      combo[idx++] = i | (SENT << 5) | (SENT << 10);
    for (unsigned i = 0; i < 16; ++i)                       // abs(analytic)
      combo[idx++] = i | (SENT << 5) | (SENT << 10) | (1u << 15);
    for (unsigned i = 0; i < 16; ++i)                       // angle(analytic)
      combo[idx++] = i | (SENT << 5) | (SENT << 10) | (2u << 15);
  }
}

// ============ K1: encoder z = x @ W_enc^T + b_enc  (WMMA f32) ============
__global__ void __launch_bounds__(256) k_encoder(const float* __restrict__ x,
                                                 const float* __restrict__ Wenc,
                                                 const float* __restrict__ benc,
                                                 float* __restrict__ zout) {
  const int lane = threadIdx.x & 31;
  const int wave = threadIdx.x >> 5;
  const int tile = blockIdx.x * 8 + wave;   // 0..2047 (16-row tiles)
  const int b = tile >> 7;
  const int t0 = (tile & 127) << 4;
  const int m = lane & 15;
  const int hi = lane >> 4;

  const float* xrow = x + (size_t)(b * Tn + t0 + m) * Fn;  // A row (M = m)
  const float* wrow = Wenc + m * Fn;                       // B col (N = m value)
  v8f c = {};
#pragma unroll
  for (int kb = 0; kb < Fn; kb += 4) {
    const int k0 = kb + 2 * hi;
    v2f a = *(const v2f*)(xrow + k0);
    v2f bb = *(const v2f*)(wrow + k0);
    c = wmma4(a, bb, c);
  }
  const float bias = benc[m];
#pragma unroll
  for (int r = 0; r < 8; ++r)
    zout[(size_t)(b * Tn + t0 + r + 8 * hi) * Ln + m] = c[r] + bias;
}

// ============ K2: parity sums  S[b][l][p] = sum_{t%2==p} z[b,t,l] ============
__global__ void __launch_bounds__(256) k_psum(const float* __restrict__ z,
                                              float* __restrict__ S) {
  __shared__ float red[256];
  const int tid = threadIdx.x;
  const int b = blockIdx.x >> 4;
  const int l = blockIdx.x & 15;
  float s = 0.0f;
  for (int t = tid; t < Tn; t += 256)           // stride 256: parity(t)==parity(tid)
    s += z[(size_t)(b * Tn + t) * Ln + l];
  red[tid] = s;
  __syncthreads();
  for (int off = 128; off >= 2; off >>= 1) {    // even offsets preserve parity
    if (tid < off) red[tid] += red[tid + off];
    __syncthreads();
  }
  if (tid == 0) {
    S[blockIdx.x * 2 + 0] = red[0];   // even-t sum
    S[blockIdx.x * 2 + 1] = red[1];   // odd-t sum
  }
}

// ===== K3: Im(analytic)[b] = Kim @ z[b], Kim[t,s] = ktab[(t-s)&2047] (WMMA) ====
__global__ void __launch_bounds__(256) k_hilbert(const float* __restrict__ z,
                                                 const float* __restrict__ ktab_g,
                                                 float* __restrict__ im) {
  __shared__ float kt[Tn];
  const int tid = threadIdx.x;
  for (int i = tid; i < Tn; i += 256) kt[i] = ktab_g[i];
  __syncthreads();

  const int lane = tid & 31, wave = tid >> 5;
  const int b = blockIdx.x >> 4;
  const int chunk = blockIdx.x & 15;
  const int t0 = chunk * 128 + wave * 16;
  const int m = lane & 15, hi = lane >> 4;
  const int row = t0 + m;
  const float* zb = z + (size_t)b * Tn * Ln;
  v8f c = {};
#pragma unroll 4
  for (int kb = 0; kb < Tn; kb += 4) {
    const int k0 = kb + 2 * hi;
    v2f a;
    a.x = kt[(row - k0) & 2047];
    a.y = kt[(row - k0 - 1) & 2047];
    v2f bb;
    bb.x = zb[(size_t)k0 * Ln + m];
    bb.y = zb[(size_t)(k0 + 1) * Ln + m];
    c = wmma4(a, bb, c);
  }
#pragma unroll
  for (int r = 0; r < 8; ++r)
    im[(size_t)(b * Tn + t0 + r + 8 * hi) * Ln + m] = c[r];
}

// ============ K4: fused theta->y_hat (K=1016) and x_hat (K=16), WMMA ============
__device__ __forceinline__ float eval_theta(const unsigned* __restrict__ combo,
                                            const float* zrow, const float* abrow,
                                            const float* anrow, int k) {
  const unsigned e = combo[k];
  const int i = e & 31;
  const int j = (e >> 5) & 31;
  const int kk = (e >> 10) & 31;
  const int src = (int)(e >> 15);
  const float* base = (src == 0) ? zrow : ((src == 1) ? abrow : anrow);
  return base[i] * zrow[j] * zrow[kk];  // index 16 -> sentinel 1.0
}

__global__ void __launch_bounds__(256) k_fused(
    const float* __restrict__ z, const float* __restrict__ im,
    const float* __restrict__ S, const unsigned* __restrict__ combo_g,
    const float* __restrict__ Wsin, const float* __restrict__ bsin,
    const float* __restrict__ Wdec, const float* __restrict__ bdec,
    float* __restrict__ yhat, float* __restrict__ xhat) {
  __shared__ unsigned combo[LIBn];
  __shared__ float tiles[8 * 3 * 16 * 17];   // per wave: z/abs/angle, stride 17
  const int tid = threadIdx.x;
  for (int i = tid; i < LIBn; i += 256) combo[i] = combo_g[i];
  __syncthreads();

  const int lane = tid & 31, wave = tid >> 5;
  const int tile = blockIdx.x * 8 + wave;
  const int b = tile >> 7;
  const int t0 = (tile & 127) << 4;
  float* zt = tiles + wave * (3 * 16 * 17);
  float* ab = zt + 16 * 17;
  float* an = ab + 16 * 17;

  const float c2N = 2.0f / (float)Tn;
  for (int idx = lane; idx < 256; idx += 32) {   // 8 iters, no divergence
    const int mm = idx >> 4, ll = idx & 15;
    const int t = t0 + mm;
    const size_t off = (size_t)(b * Tn + t) * Ln + ll;
    const float zv = z[off];
    const float iv = im[off];
    const float re = zv - c2N * S[(b * 16 + ll) * 2 + (t & 1)];
    zt[mm * 17 + ll] = zv;
    ab[mm * 17 + ll] = sqrtf(re * re + iv * iv);
    an[mm * 17 + ll] = atan2f(iv, re);
  }
  if (lane < 16) zt[lane * 17 + 16] = 1.0f;      // sentinel column = 1.0
  // same-wave LDS is in-order: no barrier needed before reads below

  const int m = lane & 15, hi = lane >> 4;
  const float* zrow = zt + m * 17;
  const float* abrow = ab + m * 17;
  const float* anrow = an + m * 17;

  // ---- y_hat = theta @ W_sindy^T + b_sindy ----
  {
    v8f c = {};
    const float* wsrow = Wsin + m * LIBn;
#pragma unroll 2
    for (int kb = 0; kb < LIBn; kb += 4) {       // 1016 = 254 * 4 exactly
      const int k0 = kb + 2 * hi;
      v2f a;
      a.x = eval_theta(combo, zrow, abrow, anrow, k0);
      a.y = eval_theta(combo, zrow, abrow, anrow, k0 + 1);
      v2f bb = *(const v2f*)(wsrow + k0);
      c = wmma4(a, bb, c);
    }
    const float bias = bsin[m];
#pragma unroll
    for (int r = 0; r < 8; ++r)
      yhat[(size_t)(b * Tn + t0 + r + 8 * hi) * Ln + m] = c[r] + bias;
  }

  // ---- x_hat = z @ W_dec^T + b_dec ----
#pragma unroll
  for (int f0 = 0; f0 < Fn; f0 += 16) {
    v8f c = {};
    const float* wdrow = Wdec + (f0 + m) * Ln;
#pragma unroll
    for (int kb = 0; kb < Ln; kb += 4) {
      const int k0 = kb + 2 * hi;
      v2f a;
      a.x = zrow[k0];
      a.y = zrow[k0 + 1];
      v2f bb = *(const v2f*)(wdrow + k0);
      c = wmma4(a, bb, c);
    }
    const float bias = bdec[f0 + m];
#pragma unroll
    for (int r = 0; r < 8; ++r)
      xhat[(size_t)(b * Tn + t0 + r + 8 * hi) * Fn + f0 + m] = c[r] + bias;
  }
}

// ======= K5: jac = broadcast(W_enc) — 256 MB nontemporal b128 stores =======
__global__ void __launch_bounds__(256) k_jac(const v4f* __restrict__ wenc4,
                                             v4f* __restrict__ jac4) {
  __shared__ v4f w[512];                      // W_enc = 2048 f32 = 512 x float4
  const int tid = threadIdx.x;
  for (int i = tid; i < 512; i += 256) w[i] = wenc4[i];
  __syncthreads();
  const size_t total = (size_t)Bn * Tn * Ln * Fn / 4;   // 16777216
  const size_t stride = (size_t)gridDim.x * 256;
  for (size_t i = (size_t)blockIdx.x * 256 + tid; i < total; i += stride)
    __builtin_nontemporal_store(w[i & 511], &jac4[i]);
}

// ============ K6: copy W_sindy / W_dec outputs ============
__global__ void k_wcopy(const float* __restrict__ Wsin,
                        const float* __restrict__ Wdec,
                        float* __restrict__ o_wsin, float* __restrict__ o_wdec) {
  const int i = blockIdx.x * 256 + threadIdx.x;
  if (i < Ln * LIBn) o_wsin[i] = Wsin[i];
  if (i < Fn * Ln) o_wdec[i] = Wdec[i];
}

extern "C" void kernel_launch(void* const* d_in, const int* in_sizes, int n_in,
                              void* d_out, int out_size, void* d_ws,
                              size_t ws_size, hipStream_t stream) {
  const float* x = (const float*)d_in[0];
  const float* Wenc = (const float*)d_in[1];
  const float* benc = (const float*)d_in[2];
  const float* Wdec = (const float*)d_in[3];
  const float* bdec = (const float*)d_in[4];
  const float* Wsin = (const float*)d_in[5];
  const float* bsin = (const float*)d_in[6];

  float* out = (float*)d_out;
  float* ws = (float*)d_ws;

  float* o_yhat = out + O_YHAT;
  float* o_xhat = out + O_XHAT;
  float* o_z = out + O_Z;          // encoder writes straight into output slot
  float* o_jac = out + O_JAC;
  float* o_wsin = out + O_WSIN;
  float* o_wdec = out + O_WDEC;

  float* w_im = ws + W_IM;
  float* w_ktab = ws + W_KTAB;
  float* w_S = ws + W_S;
  unsigned* w_combo = (unsigned*)(ws + W_COMBO);

  k_setup<<<8, 256, 0, stream>>>(w_ktab, w_combo);
  k_encoder<<<256, 256, 0, stream>>>(x, Wenc, benc, o_z);
  k_psum<<<Bn * Ln, 256, 0, stream>>>(o_z, w_S);
  k_hilbert<<<256, 256, 0, stream>>>(o_z, w_ktab, w_im);
  k_fused<<<256, 256, 0, stream>>>(o_z, w_im, w_S, w_combo, Wsin, bsin, Wdec,
                                   bdec, o_yhat, o_xhat);
  k_jac<<<8192, 256, 0, stream>>>((const v4f*)Wenc, (v4f*)o_jac);
  k_wcopy<<<64, 256, 0, stream>>>(Wsin, Wdec, o_wsin, o_wdec);
}